// GraphSAGE_29317446762711
// MI455X (gfx1250) — compile-verified
//
#include <hip/hip_runtime.h>

#define N_NODES 100000
#define N_EDGES 1600000
#define IN_CH 8
#define HID_CH 64
#define OUT_CH 2
#define ROW_TILES (N_NODES / 16)   // 6250, exact

typedef __attribute__((ext_vector_type(2))) float v2f;
typedef __attribute__((ext_vector_type(8))) float v8f;

// ---------------------------------------------------------------------------
// Stage 1: per-edge scatter of x[src] into sum1[dst] + degree count.
// 2 threads per edge (one float4 half each); atomics resolve in L2 (3.2MB).
// ---------------------------------------------------------------------------
__global__ void sage_scatter1(const int* __restrict__ ei,
                              const float* __restrict__ x,
                              float* __restrict__ deg,
                              float* __restrict__ sum1) {
    long long tid = (long long)blockIdx.x * blockDim.x + threadIdx.x;
    long long e = tid >> 1;
    int half = (int)(tid & 1);
    if (e >= N_EDGES) return;
    int s = ei[e];
    int d = ei[N_EDGES + e];
    if (half == 0) unsafeAtomicAdd(&deg[d], 1.0f);
    const float4 v = *(const float4*)(x + (long long)s * IN_CH + half * 4);
    float* dst = sum1 + (long long)d * IN_CH + half * 4;
    unsafeAtomicAdd(dst + 0, v.x);
    unsafeAtomicAdd(dst + 1, v.y);
    unsafeAtomicAdd(dst + 2, v.z);
    unsafeAtomicAdd(dst + 3, v.w);
}

// ---------------------------------------------------------------------------
// Stage 2: h = relu(mean_aggr @ Wl1^T + b1 + x @ Wr1^T), fp32 WMMA 16x16x4.
// One wave handles a 16-node row tile across all 4 column tiles (64 ch).
// A layout (16x4 f32): lanes 0-15 -> M, lane[4] selects K-pair {0,1}/{2,3}.
// B layout (4x16 f32): lanes 0-15 -> N, lane[4] selects K-pair (mirror of A).
// C layout (16x16 f32): VGPR i -> M = i + 8*lane[4], N = lane&15.
// ---------------------------------------------------------------------------
__global__ void sage_layer1(const float* __restrict__ x,
                            const float* __restrict__ sum1,
                            const float* __restrict__ deg,
                            const float* __restrict__ Wl,
                            const float* __restrict__ bl,
                            const float* __restrict__ Wr,
                            float* __restrict__ h) {
    int gwave = (int)(((long long)blockIdx.x * blockDim.x + threadIdx.x) >> 5);
    int lane = threadIdx.x & 31;
    if (gwave >= ROW_TILES) return;           // wave-uniform exit
    __builtin_prefetch(Wl, 0, 0);
    __builtin_prefetch(Wr, 0, 0);

    const int m  = lane & 15;                 // A row / B col / C col
    const int kh = lane >> 4;                 // K-pair selector
    const int k0 = kh * 2;
    const int node = gwave * 16 + m;

    const float dinv = 1.0f / fmaxf(deg[node], 1.0f);
    const float* srow = sum1 + (long long)node * IN_CH;
    const float* xrow = x    + (long long)node * IN_CH;

    v2f aA0, aA1, aX0, aX1;                   // K = 0..3 and K = 4..7
    aA0[0] = srow[k0] * dinv;      aA0[1] = srow[k0 + 1] * dinv;
    aA1[0] = srow[k0 + 4] * dinv;  aA1[1] = srow[k0 + 5] * dinv;
    aX0[0] = xrow[k0];             aX0[1] = xrow[k0 + 1];
    aX1[0] = xrow[k0 + 4];         aX1[1] = xrow[k0 + 5];

    float* hrow = h + (long long)gwave * 16 * HID_CH;

#pragma unroll
    for (int t = 0; t < 4; ++t) {
        const int n = t * 16 + m;             // output channel
        v2f bL0, bL1, bR0, bR1;               // B[k][n] = W[n][k]
        bL0[0] = Wl[n * IN_CH + k0];     bL0[1] = Wl[n * IN_CH + k0 + 1];
        bL1[0] = Wl[n * IN_CH + k0 + 4]; bL1[1] = Wl[n * IN_CH + k0 + 5];
        bR0[0] = Wr[n * IN_CH + k0];     bR0[1] = Wr[n * IN_CH + k0 + 1];
        bR1[0] = Wr[n * IN_CH + k0 + 4]; bR1[1] = Wr[n * IN_CH + k0 + 5];

        v8f c = {};
        c = __builtin_amdgcn_wmma_f32_16x16x4_f32(false, aA0, false, bL0, (short)0, c, false, false);
        c = __builtin_amdgcn_wmma_f32_16x16x4_f32(false, aA1, false, bL1, (short)0, c, false, false);
        c = __builtin_amdgcn_wmma_f32_16x16x4_f32(false, aX0, false, bR0, (short)0, c, false, false);
        c = __builtin_amdgcn_wmma_f32_16x16x4_f32(false, aX1, false, bR1, (short)0, c, false, false);

        const float bias = bl[n];
#pragma unroll
        for (int i = 0; i < 8; ++i) {
            const int mo = i + 8 * kh;        // output row within tile
            float v = c[i] + bias;
            hrow[mo * HID_CH + n] = v > 0.0f ? v : 0.0f;
        }
    }
}

// ---------------------------------------------------------------------------
// Stage 3: per-edge scatter of h[src] (64 ch) into sum2[dst].
// 16 threads per edge, 4 channels each; sum2 (25.6MB) stays L2-resident.
// ---------------------------------------------------------------------------
__global__ void sage_scatter2(const int* __restrict__ ei,
                              const float* __restrict__ h,
                              float* __restrict__ sum2) {
    long long tid = (long long)blockIdx.x * blockDim.x + threadIdx.x;
    long long e = tid >> 4;
    int g = (int)(tid & 15);
    if (e >= N_EDGES) return;
    int s = ei[e];
    int d = ei[N_EDGES + e];
    const float4 v = *(const float4*)(h + (long long)s * HID_CH + g * 4);
    float* dst = sum2 + (long long)d * HID_CH + g * 4;
    unsafeAtomicAdd(dst + 0, v.x);
    unsafeAtomicAdd(dst + 1, v.y);
    unsafeAtomicAdd(dst + 2, v.z);
    unsafeAtomicAdd(dst + 3, v.w);
}

// ---------------------------------------------------------------------------
// Stage 4: out = mean_aggr2 @ Wl2^T + b2 + h @ Wr2^T  (K=64, N=2 padded to 16).
// Branchless zero-masking of B columns >= OUT_CH keeps EXEC all-1s for WMMA.
// ---------------------------------------------------------------------------
__global__ void sage_layer2(const float* __restrict__ h,
                            const float* __restrict__ sum2,
                            const float* __restrict__ deg,
                            const float* __restrict__ Wl,
                            const float* __restrict__ bl,
                            const float* __restrict__ Wr,
                            float* __restrict__ out) {
    int gwave = (int)(((long long)blockIdx.x * blockDim.x + threadIdx.x) >> 5);
    int lane = threadIdx.x & 31;
    if (gwave >= ROW_TILES) return;           // wave-uniform exit
    __builtin_prefetch(Wl, 0, 0);
    __builtin_prefetch(Wr, 0, 0);

    const int m  = lane & 15;
    const int kh = lane >> 4;
    const int node = gwave * 16 + m;

    const float dinv = 1.0f / fmaxf(deg[node], 1.0f);
    const float* srow = sum2 + (long long)node * HID_CH;
    const float* hrow = h    + (long long)node * HID_CH;

    const int   nc  = m < OUT_CH ? m : (OUT_CH - 1);  // clamped (no OOB loads)
    const float msk = m < OUT_CH ? 1.0f : 0.0f;       // zero pad columns >= 2

    v8f c = {};
#pragma unroll
    for (int ks = 0; ks < 16; ++ks) {
        const int base = ks * 4 + kh * 2;
        v2f aA, aH, bL, bR;
        aA[0] = srow[base] * dinv;  aA[1] = srow[base + 1] * dinv;
        aH[0] = hrow[base];         aH[1] = hrow[base + 1];
        bL[0] = Wl[nc * HID_CH + base] * msk;  bL[1] = Wl[nc * HID_CH + base + 1] * msk;
        bR[0] = Wr[nc * HID_CH + base] * msk;  bR[1] = Wr[nc * HID_CH + base + 1] * msk;
        c = __builtin_amdgcn_wmma_f32_16x16x4_f32(false, aA, false, bL, (short)0, c, false, false);
        c = __builtin_amdgcn_wmma_f32_16x16x4_f32(false, aH, false, bR, (short)0, c, false, false);
    }

    const float bias = bl[nc];
    if (m < OUT_CH) {
#pragma unroll
        for (int i = 0; i < 8; ++i) {
            const int mo = i + 8 * kh;
            out[(long long)(gwave * 16 + mo) * OUT_CH + m] = c[i] + bias;
        }
    }
}

// ---------------------------------------------------------------------------
// Launch
// ---------------------------------------------------------------------------
extern "C" void kernel_launch(void* const* d_in, const int* in_sizes, int n_in,
                              void* d_out, int out_size, void* d_ws, size_t ws_size,
                              hipStream_t stream) {
    const float* x   = (const float*)d_in[0];
    const int*   ei  = (const int*)d_in[1];
    const float* Wl1 = (const float*)d_in[2];
    const float* bl1 = (const float*)d_in[3];
    const float* Wr1 = (const float*)d_in[4];
    const float* Wl2 = (const float*)d_in[5];
    const float* bl2 = (const float*)d_in[6];
    const float* Wr2 = (const float*)d_in[7];
    float* out = (float*)d_out;

    float* ws   = (float*)d_ws;
    float* deg  = ws;                                   // [N]
    float* sum1 = deg  + N_NODES;                       // [N, 8]
    float* h    = sum1 + (size_t)N_NODES * IN_CH;       // [N, 64]
    float* sum2 = h    + (size_t)N_NODES * HID_CH;      // [N, 64]

    hipMemsetAsync(deg,  0, (size_t)N_NODES * sizeof(float), stream);
    hipMemsetAsync(sum1, 0, (size_t)N_NODES * IN_CH * sizeof(float), stream);
    hipMemsetAsync(sum2, 0, (size_t)N_NODES * HID_CH * sizeof(float), stream);

    {   // degree + layer-1 feature scatter: 2 threads/edge
        long long threads = (long long)N_EDGES * 2;
        int blocks = (int)((threads + 255) / 256);
        sage_scatter1<<<blocks, 256, 0, stream>>>(ei, x, deg, sum1);
    }
    {   // layer-1 node GEMM (WMMA): one wave per 16-node tile
        long long threads = (long long)ROW_TILES * 32;
        int blocks = (int)((threads + 255) / 256);
        sage_layer1<<<blocks, 256, 0, stream>>>(x, sum1, deg, Wl1, bl1, Wr1, h);
    }
    {   // layer-2 feature scatter: 16 threads/edge
        long long threads = (long long)N_EDGES * 16;
        int blocks = (int)((threads + 255) / 256);
        sage_scatter2<<<blocks, 256, 0, stream>>>(ei, h, sum2);
    }
    {   // layer-2 node GEMM (WMMA)
        long long threads = (long long)ROW_TILES * 32;
        int blocks = (int)((threads + 255) / 256);
        sage_layer2<<<blocks, 256, 0, stream>>>(h, sum2, deg, Wl2, bl2, Wr2, out);
    }
}